// patient_GCN_79061757985142
// MI455X (gfx1250) — compile-verified
//
#include <hip/hip_runtime.h>
#include <hip/hip_bf16.h>

// ---------------- problem constants (match reference) ----------------
enum : int {
    N_NODES  = 50000,
    N_EDGES  = 640000,
    IN_DIM   = 128,
    HID      = 128,
    G_OUT    = 128,
    AX_IN    = 64,
    AX_OUT   = 64,
    OUT_CH   = 8,
    N_GRAPHS = 512,
    Z_DIM    = G_OUT + AX_OUT   // 192
};

typedef __attribute__((ext_vector_type(2))) float v2f;
typedef __attribute__((ext_vector_type(8))) float v8f;

// ---------------------------------------------------------------------
// Degree / normalization
// ---------------------------------------------------------------------
__global__ void deg_init_kernel(float* __restrict__ deg, int n) {
    int i = blockIdx.x * blockDim.x + threadIdx.x;
    if (i < n) deg[i] = 1.0f;               // self-loop contributes 1
}

__global__ void deg_edge_kernel(const long long* __restrict__ dst,
                                float* __restrict__ deg, int nE) {
    int e = blockIdx.x * blockDim.x + threadIdx.x;
    if (e < nE) atomicAdd(&deg[(int)dst[e]], 1.0f);
}

__global__ void dinv_kernel(const float* __restrict__ deg,
                            float* __restrict__ dinv, int n) {
    int i = blockIdx.x * blockDim.x + threadIdx.x;
    if (i < n) dinv[i] = rsqrtf(fmaxf(deg[i], 1.0f));
}

// ---------------------------------------------------------------------
// f32 WMMA GEMM with LDS-staged, K-pair-interleaved weights.
//   Out[M, NT*16](ldOut) = A[M,K](ldA) @ B[K, NT*16] (+ bias)
// Block = 256 threads = 8 waves. B is staged in LDS as float2 pairs:
//   ldsB2[p*N + n] = { B[2p][n], B[2p+1][n] }
// so each WMMA B-operand is a single aligned ds_load_b64 (no repack moves).
// Each wave computes one 16-row x (NT*16)-col stripe with NT accumulator
// tiles; each A element is loaded exactly once from global (contiguous b64).
// K stepped by 4 (V_WMMA_F32_16X16X4_F32). All WMMA-guarding branches are
// wave-uniform -> EXEC is all-ones at every WMMA (ISA 7.12 requirement).
// ---------------------------------------------------------------------
template <int NT>
__global__ void wmma_gemm_f32_lds_kernel(const float* __restrict__ A,
                                         const float* __restrict__ B,
                                         const float* __restrict__ bias,
                                         float* __restrict__ Out,
                                         int M, int K, int ldA, int ldOut) {
    constexpr int N = NT * 16;
    extern __shared__ float2 ldsB2[];        // (K/2) * N float2 pairs

    // cooperative, interleaving load of B into LDS
    {
        const int total = (K >> 1) * N;      // pair count
        for (int i = threadIdx.x; i < total; i += blockDim.x) {
            const int p = i / N;             // K-pair index
            const int n = i - p * N;
            float2 v;
            v.x = B[(size_t)(2 * p)     * N + n];
            v.y = B[(size_t)(2 * p + 1) * N + n];
            ldsB2[i] = v;
        }
    }
    __syncthreads();

    const int lane = threadIdx.x & 31;
    const int wv   = threadIdx.x >> 5;                     // 0..7
    const int mt   = blockIdx.x * (blockDim.x >> 5) + wv;  // m-tile index

    if (mt * 16 < M) {                                     // wave-uniform
        // A-layout: lanes 0-15 -> M=lane, K-half 0; lanes 16-31 -> K-half 1
        const int am = lane & 15;
        const int kb = (lane >> 4) << 1;                   // 0 or 2
        const float* __restrict__ arow = A + (size_t)(mt * 16 + am) * (size_t)ldA;
        const int    cl   = lane & 15;
        const float2* __restrict__ bbase = ldsB2 + cl;

        v8f acc[NT];
#pragma unroll
        for (int t = 0; t < NT; ++t) acc[t] = (v8f){};

        for (int k = 0; k < K; k += 4) {
            v2f a;
            a.x = arow[k + kb];
            a.y = arow[k + kb + 1];
            const float2* brow = bbase + ((k + kb) >> 1) * N;
#pragma unroll
            for (int t = 0; t < NT; ++t) {
                const float2 bp = brow[t * 16];            // single ds_load_b64
                v2f b; b.x = bp.x; b.y = bp.y;
                acc[t] = __builtin_amdgcn_wmma_f32_16x16x4_f32(
                    /*neg_a=*/false, a, /*neg_b=*/false, b,
                    /*c_mod=*/(short)0, acc[t],
                    /*reuse_a=*/false, /*reuse_b=*/false);
            }
        }

        // C/D layout: VGPR v -> row v + 8*(lane>=16), col lane&15
        const int rbase = mt * 16 + ((lane >> 4) << 3);
#pragma unroll
        for (int t = 0; t < NT; ++t) {
            const int c  = t * 16 + cl;
            const float bv = bias ? bias[c] : 0.0f;
#pragma unroll
            for (int v = 0; v < 8; ++v)
                Out[(size_t)(rbase + v) * (size_t)ldOut + c] = acc[t][v] + bv;
        }
    }
}

// ---------------------------------------------------------------------
// GCN aggregation: agg[i,f] = b[f] + dinv[i]^2 * h[i,f]  (self-loop + bias)
// then per-edge scatter: agg[dst,f] += dinv[src]*dinv[dst]*h[src,f]
// ---------------------------------------------------------------------
__global__ void agg_init_kernel(float* __restrict__ agg,
                                const float* __restrict__ h,
                                const float* __restrict__ dinv,
                                const float* __restrict__ bias,
                                int n, int F) {
    int idx = blockIdx.x * blockDim.x + threadIdx.x;
    if (idx >= n * F) return;
    int i = idx / F, f = idx - i * F;
    float di = dinv[i];
    agg[idx] = bias[f] + di * di * h[idx];
}

// one wave per edge; lane handles 4 consecutive features (F == 128)
__global__ void edge_agg_kernel(float* __restrict__ agg,
                                const float* __restrict__ h,
                                const long long* __restrict__ src,
                                const long long* __restrict__ dst,
                                const float* __restrict__ dinv, int nE) {
    int e = (blockIdx.x * blockDim.x + threadIdx.x) >> 5;
    if (e >= nE) return;
    int lane = threadIdx.x & 31;
    int s = (int)src[e];
    int d = (int)dst[e];
    float nrm = dinv[s] * dinv[d];
    const float4 hv = ((const float4*)(h + (size_t)s * HID))[lane];
    float* ad = agg + (size_t)d * HID + lane * 4;
    atomicAdd(ad + 0, nrm * hv.x);
    atomicAdd(ad + 1, nrm * hv.y);
    atomicAdd(ad + 2, nrm * hv.z);
    atomicAdd(ad + 3, nrm * hv.w);
}

__global__ void relu_kernel(float* __restrict__ x, int n) {
    int i = blockIdx.x * blockDim.x + threadIdx.x;
    if (i < n) x[i] = fmaxf(x[i], 0.0f);
}

// ---------------------------------------------------------------------
// global mean pool
// ---------------------------------------------------------------------
__global__ void pool_init_kernel(float* __restrict__ sums,
                                 float* __restrict__ cnts) {
    int i = blockIdx.x * blockDim.x + threadIdx.x;
    if (i < N_GRAPHS * HID) sums[i] = 0.0f;
    if (i < N_GRAPHS) cnts[i] = 0.0f;
}

// one wave per node
__global__ void pool_accum_kernel(float* __restrict__ sums,
                                  float* __restrict__ cnts,
                                  const float* __restrict__ h,
                                  const long long* __restrict__ batch, int n) {
    int i = (blockIdx.x * blockDim.x + threadIdx.x) >> 5;
    if (i >= n) return;
    int lane = threadIdx.x & 31;
    int g = (int)batch[i];
    const float4 hv = ((const float4*)(h + (size_t)i * HID))[lane];
    float* sg = sums + (size_t)g * HID + lane * 4;
    atomicAdd(sg + 0, hv.x);
    atomicAdd(sg + 1, hv.y);
    atomicAdd(sg + 2, hv.z);
    atomicAdd(sg + 3, hv.w);
    if (lane == 0) atomicAdd(&cnts[g], 1.0f);
}

__global__ void pool_div_kernel(float* __restrict__ pooled,
                                const float* __restrict__ sums,
                                const float* __restrict__ cnts) {
    int idx = blockIdx.x * blockDim.x + threadIdx.x;
    if (idx >= N_GRAPHS * HID) return;
    int g = idx / HID;
    pooled[idx] = sums[idx] / fmaxf(cnts[g], 1.0f);
}

// ---------------------------------------------------------------------
// final head: out[g,o] = lin2_b[o] + sum_k z[g,k] * lin2_W[k,o]   (K=192,N=8)
// ---------------------------------------------------------------------
__global__ void final_kernel(float* __restrict__ out,
                             const float* __restrict__ z,
                             const float* __restrict__ W,
                             const float* __restrict__ b) {
    int idx = blockIdx.x * blockDim.x + threadIdx.x;
    if (idx >= N_GRAPHS * OUT_CH) return;
    int g = idx / OUT_CH, o = idx - g * OUT_CH;
    const float* zr = z + (size_t)g * Z_DIM;
    float acc = b[o];
    for (int k = 0; k < Z_DIM; ++k)
        acc += zr[k] * W[k * OUT_CH + o];
    out[idx] = acc;
}

// ---------------------------------------------------------------------
extern "C" void kernel_launch(void* const* d_in, const int* in_sizes, int n_in,
                              void* d_out, int out_size, void* d_ws, size_t ws_size,
                              hipStream_t stream) {
    const float*     x      = (const float*)d_in[0];
    const long long* eidx   = (const long long*)d_in[1];   // int64 [2,E]
    const long long* batch  = (const long long*)d_in[2];   // int64 [N]
    const float*     ax     = (const float*)d_in[3];
    const float*     W1     = (const float*)d_in[4];
    const float*     b1     = (const float*)d_in[5];
    const float*     W2     = (const float*)d_in[6];
    const float*     b2     = (const float*)d_in[7];
    const float*     lin1W  = (const float*)d_in[8];
    const float*     lin1b  = (const float*)d_in[9];
    const float*     axW    = (const float*)d_in[10];
    const float*     axb    = (const float*)d_in[11];
    const float*     lin2W  = (const float*)d_in[12];
    const float*     lin2b  = (const float*)d_in[13];
    float*           out    = (float*)d_out;

    const long long* src = eidx;            // row 0
    const long long* dst = eidx + N_EDGES;  // row 1

    // workspace layout (floats)
    float* ws     = (float*)d_ws;
    float* deg    = ws;                                   // N
    float* dinv   = deg + N_NODES;                        // N
    float* bufA   = dinv + N_NODES;                       // N*HID  (GEMM out)
    float* bufB   = bufA + (size_t)N_NODES * HID;         // N*HID  (agg/h)
    float* sums   = bufB + (size_t)N_NODES * HID;         // G*HID
    float* cnts   = sums + N_GRAPHS * HID;                // G
    float* pooled = cnts + N_GRAPHS;                      // G*HID
    float* zbuf   = pooled + N_GRAPHS * HID;              // G*Z_DIM
    (void)in_sizes; (void)n_in; (void)out_size; (void)ws_size;

    const int T = 256;              // 8 waves per block
    auto cdiv = [](long long a, long long b) { return (int)((a + b - 1) / b); };

    // ---- normalization --------------------------------------------------
    deg_init_kernel<<<cdiv(N_NODES, T), T, 0, stream>>>(deg, N_NODES);
    deg_edge_kernel<<<cdiv(N_EDGES, T), T, 0, stream>>>(dst, deg, N_EDGES);
    dinv_kernel<<<cdiv(N_NODES, T), T, 0, stream>>>(deg, dinv, N_NODES);

    const int mtilesN = N_NODES / 16;        // 3125
    const int mtilesG = N_GRAPHS / 16;       // 32
    const size_t lds128 = (size_t)HID   * 128 * sizeof(float);  // 64 KB
    const size_t lds64  = (size_t)AX_IN * 64  * sizeof(float);  // 16 KB

    // ---- layer 1 --------------------------------------------------------
    wmma_gemm_f32_lds_kernel<8><<<cdiv(mtilesN, 8), T, lds128, stream>>>(
        x, W1, nullptr, bufA, N_NODES, IN_DIM, IN_DIM, HID);
    agg_init_kernel<<<cdiv((long long)N_NODES * HID, T), T, 0, stream>>>(
        bufB, bufA, dinv, b1, N_NODES, HID);
    edge_agg_kernel<<<cdiv((long long)N_EDGES * 32, T), T, 0, stream>>>(
        bufB, bufA, src, dst, dinv, N_EDGES);
    relu_kernel<<<cdiv((long long)N_NODES * HID, T), T, 0, stream>>>(
        bufB, N_NODES * HID);

    // ---- layer 2 --------------------------------------------------------
    wmma_gemm_f32_lds_kernel<8><<<cdiv(mtilesN, 8), T, lds128, stream>>>(
        bufB, W2, nullptr, bufA, N_NODES, HID, HID, HID);
    agg_init_kernel<<<cdiv((long long)N_NODES * HID, T), T, 0, stream>>>(
        bufB, bufA, dinv, b2, N_NODES, HID);
    edge_agg_kernel<<<cdiv((long long)N_EDGES * 32, T), T, 0, stream>>>(
        bufB, bufA, src, dst, dinv, N_EDGES);
    relu_kernel<<<cdiv((long long)N_NODES * HID, T), T, 0, stream>>>(
        bufB, N_NODES * HID);

    // ---- mean pool ------------------------------------------------------
    pool_init_kernel<<<cdiv(N_GRAPHS * HID, T), T, 0, stream>>>(sums, cnts);
    pool_accum_kernel<<<cdiv((long long)N_NODES * 32, T), T, 0, stream>>>(
        sums, cnts, bufB, batch, N_NODES);
    pool_div_kernel<<<cdiv(N_GRAPHS * HID, T), T, 0, stream>>>(pooled, sums, cnts);

    // ---- heads: write both GEMMs straight into concatenated z ----------
    wmma_gemm_f32_lds_kernel<8><<<cdiv(mtilesG, 8), T, lds128, stream>>>(
        pooled, lin1W, lin1b, zbuf, N_GRAPHS, HID, HID, Z_DIM);
    wmma_gemm_f32_lds_kernel<4><<<cdiv(mtilesG, 8), T, lds64, stream>>>(
        ax, axW, axb, zbuf + G_OUT, N_GRAPHS, AX_IN, AX_IN, Z_DIM);

    // ---- final projection ----------------------------------------------
    final_kernel<<<cdiv(N_GRAPHS * OUT_CH, T), T, 0, stream>>>(
        out, zbuf, lin2W, lin2b);
}